// HOG_68083821576720
// MI455X (gfx1250) — compile-verified
//
#include <hip/hip_runtime.h>

// ---------------------------------------------------------------------------
// HOG, fused single kernel for MI455X (gfx1250).
//   grid = 128 workgroups (1 per image), 1024 threads (32 wave32 waves).
//   Phase 0: TDM (tensor_load_to_lds) DMAs the whole 256x256 fp32 image
//            into LDS, 8 rows per wave (32 concurrent descriptors).
//   Phase 1: 1 thread per 8x8 cell -> 9-bin histogram (LDS cells).
//   Phase 2: 1 thread per 2x2 block -> rsqrt(sum of 36 squares + eps).
//   Phase 3: coalesced normalized output, matching reference flat layout.
// ---------------------------------------------------------------------------

#define HW      256
#define HC      32          // cells per side
#define HB      31          // blocks per side
#define NBINS   9
#define BLK_LEN 36          // 2*2*9
#define OUT_PER_IMG (HB*HB*BLK_LEN)   // 34596
#define HOG_EPS 1e-12f
#define PI_F    3.14159265358979323846f

typedef __attribute__((ext_vector_type(4))) unsigned int u32x4;
typedef __attribute__((ext_vector_type(8))) int          i32x8;
typedef __attribute__((ext_vector_type(4))) int          i32x4;

// LDS layout (in floats)
#define IMG_OFF  0
#define CELL_OFF (HW*HW)                       // 65536
#define NORM_OFF (CELL_OFF + HC*HC*NBINS)      // 65536 + 9216
#define SMEM_FLOATS (NORM_OFF + HB*HB)         // + 961  -> 75713 floats (~296 KB)

extern "C" __global__ __launch_bounds__(1024, 1)
void hog_fused_kernel(const float* __restrict__ x, float* __restrict__ out)
{
    extern __shared__ float smem[];
    float* img   = smem + IMG_OFF;
    float* cells = smem + CELL_OFF;
    float* invn  = smem + NORM_OFF;

    const unsigned tid = threadIdx.x;
    const unsigned b   = blockIdx.x;

    // ---------------- Phase 0: TDM image DMA into LDS -----------------------
    // Each of the 32 waves DMAs an 8-row x 256-col fp32 slice of the image.
    {
        const unsigned wid = __builtin_amdgcn_readfirstlane(tid >> 5);   // 0..31, SGPR
        const unsigned row0 = wid * 8u;

        unsigned long long ga =
            (unsigned long long)(const void*)x +
            ((unsigned long long)b * (HW * HW) + (unsigned long long)row0 * HW) * 4ull;
        unsigned lds_byte = (unsigned)(unsigned long long)(void*)img + row0 * HW * 4u;

        // D# group 0: count=1 | lds_addr | global_addr | type=2
        u32x4 g0;
        g0[0] = 1u;                                   // count = 1 (valid descriptor)
        g0[1] = lds_byte;                             // lds_addr (bytes)
        g0[2] = (unsigned)ga;                         // global_addr[31:0]
        g0[3] = (unsigned)(ga >> 32) | (2u << 30);    // global_addr[56:32] | type=2

        // D# group 1: data_size=4B, tensor 256 x 8, tile 256 x 8, stride0 = 256
        i32x8 g1;
        g1[0] = (int)(2u << 16);      // wg_mask=0 | data_size=2 (4 bytes)
        g1[1] = (int)(HW  << 16);     // [63:48]  = tensor_dim0 lo16 = 256
        g1[2] = (int)(8u  << 16);     // [95:80]  = tensor_dim1 lo16 = 8 rows
        g1[3] = (int)(HW  << 16);     // [127:112]= tile_dim0       = 256
        g1[4] = (int)8;               // [143:128]= tile_dim1 = 8 ; tile_dim2 = 0
        g1[5] = (int)HW;              // tensor_dim0_stride[31:0] = 256 elements
        g1[6] = 0;                    // stride0 hi | tensor_dim1_stride lo = 0
        g1[7] = 0;

        i32x4 gz4 = {0, 0, 0, 0};     // groups 2/3 unused (2D tensor)
        i32x8 gz8 = {0, 0, 0, 0, 0, 0, 0, 0};

        __builtin_amdgcn_tensor_load_to_lds(g0, g1, gz4, gz4, gz8, 0);
        __builtin_amdgcn_s_wait_tensorcnt(0);         // wave's DMA complete
    }
    __syncthreads();                                  // all slices visible

    // ---------------- Phase 1: per-cell 9-bin histograms --------------------
    {
        const int ch = (int)(tid >> 5);               // cell row 0..31
        const int cw = (int)(tid & 31);               // cell col 0..31
        const int r0 = ch * 8, c0 = cw * 8;

        float acc[NBINS];
        #pragma unroll
        for (int k = 0; k < NBINS; ++k) acc[k] = 0.0f;

        for (int dr = 0; dr < 8; ++dr) {
            const int r  = r0 + dr;
            const int rm = (r == 0)      ? 0       : r - 1;   // 'edge' pad
            const int rp = (r == HW - 1) ? HW - 1  : r + 1;
            #pragma unroll
            for (int dc = 0; dc < 8; ++dc) {
                const int c  = c0 + dc;
                const int cm = (c == 0)      ? 0      : c - 1;
                const int cp = (c == HW - 1) ? HW - 1 : c + 1;

                const float gx = img[r  * HW + cp] - img[r  * HW + cm];
                const float gy = img[rp * HW + c ] - img[rm * HW + c ];
                const float mag = sqrtf(gx * gx + gy * gy + HOG_EPS);

                float a = atan2f(gy, gx);             // [-pi, pi]
                if (a < 0.0f) a += PI_F;              // mod pi -> [0, pi)
                int bin = (int)(a * ((float)NBINS / PI_F));
                bin = bin < 0 ? 0 : (bin > NBINS - 1 ? NBINS - 1 : bin);

                #pragma unroll
                for (int k = 0; k < NBINS; ++k)       // branchless one-hot vote
                    acc[k] += (k == bin) ? mag : 0.0f;
            }
        }
        #pragma unroll
        for (int k = 0; k < NBINS; ++k)
            cells[tid * NBINS + k] = acc[k];
    }
    __syncthreads();

    // ---------------- Phase 2: per-block inverse L2 norms -------------------
    if (tid < HB * HB) {
        const int bh = (int)tid / HB;
        const int bw = (int)tid - bh * HB;
        float s = 0.0f;
        #pragma unroll
        for (int ho = 0; ho < 2; ++ho)
            #pragma unroll
            for (int wo = 0; wo < 2; ++wo) {
                const float* cp_ = &cells[((bh + ho) * HC + (bw + wo)) * NBINS];
                #pragma unroll
                for (int k = 0; k < NBINS; ++k) { const float v = cp_[k]; s += v * v; }
            }
        invn[tid] = rsqrtf(s + HOG_EPS);              // 1 / sqrt(sum + eps)
    }
    __syncthreads();

    // ---------------- Phase 3: coalesced normalized output ------------------
    // flat j within a block = (hoff*2 + woff)*9 + bin  (reference transpose 0,1,3,2,4,5)
    float* outp = out + (unsigned long long)b * OUT_PER_IMG;
    for (int i = (int)tid; i < OUT_PER_IMG; i += 1024) {
        const int blk = i / BLK_LEN;
        const int j   = i - blk * BLK_LEN;
        const int bh  = blk / HB;
        const int bw  = blk - bh * HB;
        const int j9  = j / 9;
        const int ho  = j9 >> 1;
        const int wo  = j9 & 1;
        const int k   = j - j9 * 9;
        outp[i] = cells[((bh + ho) * HC + (bw + wo)) * NBINS + k] * invn[blk];
    }
}

extern "C" void kernel_launch(void* const* d_in, const int* in_sizes, int n_in,
                              void* d_out, int out_size, void* d_ws, size_t ws_size,
                              hipStream_t stream)
{
    (void)n_in; (void)out_size; (void)d_ws; (void)ws_size;
    const float* x = (const float*)d_in[0];
    float* out = (float*)d_out;
    const int B = in_sizes[0] / (HW * HW);            // 128 images
    const size_t smem_bytes = (size_t)SMEM_FLOATS * sizeof(float);
    hipLaunchKernelGGL(hog_fused_kernel, dim3(B), dim3(1024), smem_bytes, stream,
                       x, out);
}